// BERTVARActQuantizer_47691316855354
// MI455X (gfx1250) — compile-verified
//
#include <hip/hip_runtime.h>
#include <hip/hip_bf16.h>

#define QMAX_ITERS 10
#define QNBLK 1024
#define QTPB 256
#define QBIG 3.0e38f

typedef float v2f_t __attribute__((ext_vector_type(2)));
typedef float v8f_t __attribute__((ext_vector_type(8)));

// Cross-kernel device state living in d_ws. All iteration scalars stay on
// device so the whole pipeline is graph-capturable with zero host readback.
struct QState {
  float thrs[QMAX_ITERS];
  float deltas[QMAX_ITERS];
  float zps[QMAX_ITERS];
  float valids[QMAX_ITERS];   // 1.0 / 0.0
  float sLo;                  // band lower bound = thr[i-2] (0 initially)
  float sHi;                  // band upper bound = thr[i-1] (0 initially)
  unsigned skip;              // 1 => outlier set empty, remaining iters inactive
  float pMaxAbs;              // stats of previous iteration's active set
  float pMinx;
  float pMaxx;
  unsigned pInactive;         // #elements already zeroed (incl. original zeros)
};

struct QPartials {
  float bandMin[QNBLK];
  float bandMax[QNBLK];
  float sum[QNBLK];
  float sumsq[QNBLK];
  float maxA[QNBLK];
  float minX[QNBLK];
  float maxX[QNBLK];
  unsigned cnt[QNBLK];
};

__device__ __forceinline__ float wshfl_xor(float v, int m) { return __shfl_xor(v, m, 32); }

__global__ void q_init(QState* st) {
  if (threadIdx.x == 0) {
    st->sLo = 0.0f; st->sHi = 0.0f; st->skip = 0u;
    st->pMaxAbs = 0.0f; st->pMinx = QBIG; st->pMaxx = -QBIG; st->pInactive = 0u;
  }
}

// One fused streaming pass:
//   stats over {|x| > hi}:  n, sum|x|, sum x^2, max|x|, min x, max x
//   band min/max over {lo < |x| <= hi}   (inlier candidates of previous iter)
__global__ __launch_bounds__(QTPB) void q_pass(const float* __restrict__ x, long N,
                                               const QState* __restrict__ st,
                                               QPartials* __restrict__ pt) {
  if (st->skip) return;  // uniform: whole pass becomes a no-op once done
  const float lo = st->sLo;
  const float hi = st->sHi;
  float bmin = QBIG, bmax = -QBIG, s = 0.0f, ss = 0.0f;
  float ma = 0.0f, mn = QBIG, mx = -QBIG;
  unsigned c = 0u;

  const long n4 = N >> 2;
  const float4* __restrict__ x4 = (const float4*)x;
  const long stride = (long)gridDim.x * QTPB;
  for (long i = (long)blockIdx.x * QTPB + threadIdx.x; i < n4; i += stride) {
    __builtin_prefetch((const void*)(x4 + i + stride), 0, 3);  // global_prefetch_b8
    float4 v = x4[i];
    float e0[4] = {v.x, v.y, v.z, v.w};
#pragma unroll
    for (int k = 0; k < 4; ++k) {
      float xv = e0[k];
      float a = fabsf(xv);
      if (a > hi) {
        c += 1u;
        s += a;
        ss = fmaf(a, a, ss);
        ma = fmaxf(ma, a);
        mn = fminf(mn, xv);
        mx = fmaxf(mx, xv);
      } else if (a > lo) {
        bmin = fminf(bmin, xv);
        bmax = fmaxf(bmax, xv);
      }
    }
  }
  // scalar tail (N not divisible by 4)
  for (long i = (n4 << 2) + (long)blockIdx.x * QTPB + threadIdx.x; i < N; i += stride) {
    float xv = x[i];
    float a = fabsf(xv);
    if (a > hi) {
      c += 1u; s += a; ss = fmaf(a, a, ss);
      ma = fmaxf(ma, a); mn = fminf(mn, xv); mx = fmaxf(mx, xv);
    } else if (a > lo) {
      bmin = fminf(bmin, xv); bmax = fmaxf(bmax, xv);
    }
  }

  // wave32 butterfly reduction
#pragma unroll
  for (int m = 16; m; m >>= 1) {
    bmin = fminf(bmin, wshfl_xor(bmin, m));
    bmax = fmaxf(bmax, wshfl_xor(bmax, m));
    s += wshfl_xor(s, m);
    ss += wshfl_xor(ss, m);
    ma = fmaxf(ma, wshfl_xor(ma, m));
    mn = fminf(mn, wshfl_xor(mn, m));
    mx = fmaxf(mx, wshfl_xor(mx, m));
    c += (unsigned)__shfl_xor((int)c, m, 32);
  }
  __shared__ float sBMin[8], sBMax[8], sS[8], sSS[8], sMA[8], sMN[8], sMX[8];
  __shared__ unsigned sC[8];
  const int wave = threadIdx.x >> 5;
  const int lane = threadIdx.x & 31;
  if (lane == 0) {
    sBMin[wave] = bmin; sBMax[wave] = bmax; sS[wave] = s; sSS[wave] = ss;
    sMA[wave] = ma; sMN[wave] = mn; sMX[wave] = mx; sC[wave] = c;
  }
  __syncthreads();
  if (threadIdx.x == 0) {
    for (int w = 1; w < QTPB / 32; ++w) {
      bmin = fminf(bmin, sBMin[w]); bmax = fmaxf(bmax, sBMax[w]);
      s += sS[w]; ss += sSS[w];
      ma = fmaxf(ma, sMA[w]); mn = fminf(mn, sMN[w]); mx = fmaxf(mx, sMX[w]);
      c += sC[w];
    }
    pt->bandMin[blockIdx.x] = bmin; pt->bandMax[blockIdx.x] = bmax;
    pt->sum[blockIdx.x] = s; pt->sumsq[blockIdx.x] = ss;
    pt->maxA[blockIdx.x] = ma; pt->minX[blockIdx.x] = mn; pt->maxX[blockIdx.x] = mx;
    pt->cnt[blockIdx.x] = c;
  }
}

// Single-workgroup finalize. Sum / sum-of-squares partials are reduced with
// chained V_WMMA_F32_16X16X4_F32 (B = ones => each WMMA folds 64 values into
// the f32 accumulator tile); min/max/count use shuffle trees.
__global__ __launch_bounds__(256) void q_finalize(QState* __restrict__ st,
                                                  const QPartials* __restrict__ pt,
                                                  int iter, unsigned Ntotal) {
  __shared__ float sSum[256], sSq[256];
  __shared__ float sBMin[8], sBMax[8], sMA[8], sMN[8], sMX[8];
  __shared__ unsigned sC[8];
  const int t = threadIdx.x;
  const unsigned skip = st->skip;

  float bmin = QBIG, bmax = -QBIG, ma = 0.0f, mn = QBIG, mx = -QBIG;
  unsigned c = 0u;
  float totS = 0.0f, totQ = 0.0f;

  if (!skip) {
    float s = 0.0f, ss = 0.0f;
    for (int i = t; i < QNBLK; i += 256) {
      bmin = fminf(bmin, pt->bandMin[i]); bmax = fmaxf(bmax, pt->bandMax[i]);
      s += pt->sum[i]; ss += pt->sumsq[i];
      ma = fmaxf(ma, pt->maxA[i]); mn = fminf(mn, pt->minX[i]); mx = fmaxf(mx, pt->maxX[i]);
      c += pt->cnt[i];
    }
    sSum[t] = s; sSq[t] = ss;
#pragma unroll
    for (int m = 16; m; m >>= 1) {
      bmin = fminf(bmin, wshfl_xor(bmin, m));
      bmax = fmaxf(bmax, wshfl_xor(bmax, m));
      ma = fmaxf(ma, wshfl_xor(ma, m));
      mn = fminf(mn, wshfl_xor(mn, m));
      mx = fmaxf(mx, wshfl_xor(mx, m));
      c += (unsigned)__shfl_xor((int)c, m, 32);
    }
    const int wave = t >> 5;
    const int lane = t & 31;
    if (lane == 0) {
      sBMin[wave] = bmin; sBMax[wave] = bmax; sMA[wave] = ma;
      sMN[wave] = mn; sMX[wave] = mx; sC[wave] = c;
    }
    __syncthreads();
    if (t < 32) {  // wave 0, EXEC all ones -> WMMA legal
      v2f_t bones = {1.0f, 1.0f};
      v8f_t acc = {0.f, 0.f, 0.f, 0.f, 0.f, 0.f, 0.f, 0.f};
      v8f_t acc2 = {0.f, 0.f, 0.f, 0.f, 0.f, 0.f, 0.f, 0.f};
#pragma unroll
      for (int j = 0; j < 4; ++j) {
        v2f_t a  = { sSum[t + 64 * j], sSum[t + 64 * j + 32] };
        v2f_t a2 = { sSq [t + 64 * j], sSq [t + 64 * j + 32] };
        acc  = __builtin_amdgcn_wmma_f32_16x16x4_f32(false, a,  false, bones, (short)0, acc,  false, false);
        acc2 = __builtin_amdgcn_wmma_f32_16x16x4_f32(false, a2, false, bones, (short)0, acc2, false, false);
      }
      float ps = acc[0] + acc[1] + acc[2] + acc[3] + acc[4] + acc[5] + acc[6] + acc[7];
      float pq = acc2[0] + acc2[1] + acc2[2] + acc2[3] + acc2[4] + acc2[5] + acc2[6] + acc2[7];
      ps += wshfl_xor(ps, 16);  // column 0 lives in lanes 0 (M0..7) and 16 (M8..15)
      pq += wshfl_xor(pq, 16);
      totS = ps; totQ = pq;
      if (t == 0) {
        for (int w = 1; w < 8; ++w) {
          bmin = fminf(bmin, sBMin[w]); bmax = fmaxf(bmax, sBMax[w]);
          ma = fmaxf(ma, sMA[w]); mn = fminf(mn, sMN[w]); mx = fmaxf(mx, sMX[w]);
          c += sC[w];
        }
      }
    }
  }

  if (t == 0) {
    const bool sk = (skip != 0u);
    if (iter >= 1) {
      if (!sk) {  // valid[iter-1] == !skip: record delta/zp of iteration iter-1
        float thrP = st->sHi;
        float xmin, xmax;
        if (thrP > st->pMaxAbs) {  // inliers = non-zero elements of current vec
          xmin = st->pMinx; xmax = st->pMaxx;
        } else {                   // inliers = band, plus 0 if anything zeroed
          xmin = bmin; xmax = bmax;
          if (st->pInactive > 0u) { xmin = fminf(xmin, 0.0f); xmax = fmaxf(xmax, 0.0f); }
        }
        float delta = (xmax - xmin) / 255.0f;
        st->deltas[iter - 1] = delta;
        st->zps[iter - 1] = rintf(-xmin / delta);
      } else {  // invalid group: never referenced (gated by valids), write benign
        st->deltas[iter - 1] = 1.0f;
        st->zps[iter - 1] = 0.0f;
      }
    }
    if (iter < QMAX_ITERS) {
      if (sk) {
        st->thrs[iter] = 0.0f;
        st->valids[iter] = 0.0f;
      } else {
        const unsigned n = c;
        const float nf = fmaxf((float)n, 1.0f);
        const float mean = totS / nf;
        const float var = fmaxf(totQ - totS * totS / nf, 0.0f) / fmaxf(nf - 1.0f, 1.0f);
        const float thr = mean + 3.0f * sqrtf(var);
        st->thrs[iter] = thr;
        st->valids[iter] = (n > 0u) ? 1.0f : 0.0f;
        st->pMaxAbs = ma; st->pMinx = mn; st->pMaxx = mx;
        st->pInactive = Ntotal - n;
        st->sLo = st->sHi;
        st->sHi = thr;
        st->skip = (n > 0u) ? 0u : 1u;
      }
    }
  }
}

__global__ __launch_bounds__(QTPB) void q_dequant(const float* __restrict__ x,
                                                  float* __restrict__ out, long N,
                                                  const QState* __restrict__ st) {
  // Uniform state -> scalar loads, broadcast to all lanes.
  float thr[QMAX_ITERS], val[QMAX_ITERS], del[QMAX_ITERS], zp[QMAX_ITERS];
#pragma unroll
  for (int i = 0; i < QMAX_ITERS; ++i) {
    thr[i] = st->thrs[i]; val[i] = st->valids[i];
    del[i] = st->deltas[i]; zp[i] = st->zps[i];
  }
  int ng = 0;
#pragma unroll
  for (int i = 0; i < QMAX_ITERS; ++i) ng += (val[i] != 0.0f) ? 1 : 0;
  const int gcap = (ng - 1 > 0) ? (ng - 1) : 0;

  const long n4 = N >> 2;
  const float4* __restrict__ x4 = (const float4*)x;
  float4* __restrict__ o4 = (float4*)out;
  const long stride = (long)gridDim.x * QTPB;
  for (long i = (long)blockIdx.x * QTPB + threadIdx.x; i < n4; i += stride) {
    __builtin_prefetch((const void*)(x4 + i + stride), 0, 3);
    float4 v = x4[i];
    float e0[4] = {v.x, v.y, v.z, v.w};
    float r[4];
#pragma unroll
    for (int k = 0; k < 4; ++k) {
      float xv = e0[k];
      float a = fabsf(xv);
      int gi = 0;
#pragma unroll
      for (int g = 1; g < QMAX_ITERS; ++g) {
        bool band = (a > thr[g - 1]) && (a <= thr[g]) && (val[g] != 0.0f);
        gi = band ? g : gi;
      }
      gi = (gi > gcap) ? gcap : gi;
      float d = del[0], z = zp[0];
#pragma unroll
      for (int g = 1; g < QMAX_ITERS; ++g) {
        d = (gi == g) ? del[g] : d;
        z = (gi == g) ? zp[g] : z;
      }
      float q = rintf(xv / d) + z;           // round-half-even like jnp.round
      q = (q < 0.0f) ? 0.0f : q;             // NaN-propagating clamp (jnp.clip)
      q = (q > 255.0f) ? 255.0f : q;
      r[k] = (q - z) * d;
    }
    float4 ov; ov.x = r[0]; ov.y = r[1]; ov.z = r[2]; ov.w = r[3];
    o4[i] = ov;
  }
  for (long i = (n4 << 2) + (long)blockIdx.x * QTPB + threadIdx.x; i < N; i += stride) {
    float xv = x[i];
    float a = fabsf(xv);
    int gi = 0;
#pragma unroll
    for (int g = 1; g < QMAX_ITERS; ++g) {
      bool band = (a > thr[g - 1]) && (a <= thr[g]) && (val[g] != 0.0f);
      gi = band ? g : gi;
    }
    gi = (gi > gcap) ? gcap : gi;
    float d = del[0], z = zp[0];
#pragma unroll
    for (int g = 1; g < QMAX_ITERS; ++g) {
      d = (gi == g) ? del[g] : d;
      z = (gi == g) ? zp[g] : z;
    }
    float q = rintf(xv / d) + z;
    q = (q < 0.0f) ? 0.0f : q;
    q = (q > 255.0f) ? 255.0f : q;
    out[i] = (q - z) * d;
  }
}

extern "C" void kernel_launch(void* const* d_in, const int* in_sizes, int n_in,
                              void* d_out, int out_size, void* d_ws, size_t ws_size,
                              hipStream_t stream) {
  (void)n_in; (void)out_size; (void)ws_size;
  const float* x = (const float*)d_in[0];
  const long N = (long)in_sizes[0];

  QState* st = (QState*)d_ws;
  QPartials* pt = (QPartials*)((char*)d_ws + ((sizeof(QState) + 255) & ~(size_t)255));

  q_init<<<1, 32, 0, stream>>>(st);
  // Iteration 0 stats pass: lo = hi = 0 -> stats over {|x| > 0}, empty band.
  q_pass<<<QNBLK, QTPB, 0, stream>>>(x, N, st, pt);
  q_finalize<<<1, 256, 0, stream>>>(st, pt, 0, (unsigned)N);
  // Fused passes: band for iter i-1 (delta/zp) + stats for iter i (thr).
  for (int it = 1; it <= QMAX_ITERS; ++it) {
    q_pass<<<QNBLK, QTPB, 0, stream>>>(x, N, st, pt);
    q_finalize<<<1, 256, 0, stream>>>(st, pt, it, (unsigned)N);
  }
  q_dequant<<<4096, QTPB, 0, stream>>>(x, (float*)d_out, N, st);
}